// OracleLayer_58918361367154
// MI455X (gfx1250) — compile-verified
//
#include <hip/hip_runtime.h>
#include <stdint.h>

#define LDS_AS __attribute__((address_space(3)))
#define GLB_AS __attribute__((address_space(1)))

typedef int v4i __attribute__((__vector_size__(4 * sizeof(int))));

#ifndef __has_builtin
#define __has_builtin(x) 0
#endif

#if __has_builtin(__builtin_amdgcn_global_load_async_to_lds_b128)
#define ORACLE_USE_ASYNC 1
#else
#define ORACLE_USE_ASYNC 0
#endif

#if __has_builtin(__builtin_amdgcn_s_wait_asynccnt)
#define WAIT_ASYNCCNT(n) __builtin_amdgcn_s_wait_asynccnt(n)
#else
#define WAIT_ASYNCCNT(n) asm volatile("s_wait_asynccnt %0" ::"i"(n) : "memory")
#endif

constexpr int kThreads = 256;   // 8 wave32s per block
constexpr int kWaves   = kThreads / 32;
constexpr int kE       = 4;
constexpr int kDepth   = 4;     // async-to-LDS pipeline depth (16 KB LDS ring)

__global__ __launch_bounds__(kThreads) void oracle_effective_kernel(
    const float* __restrict__ g0, const float* __restrict__ g1,
    const float* __restrict__ g2, const float* __restrict__ g3,
    const int* __restrict__ labels, float* __restrict__ out, int L) {
  const int b = blockIdx.x;
  const int tid = threadIdx.x;
  const int vecRow = L >> 2;                                   // float4 per row
  const int nchunks = (vecRow + kThreads - 1) / kThreads;

  __shared__ float4 tile[kDepth][kThreads];
  __shared__ float  red_v[kWaves];
  __shared__ int    red_i[kWaves];
  __shared__ int    s_pred[kE];
  __shared__ float  s_lab[kE];
  __shared__ int    s_best;

  const int label = labels[b];

#pragma unroll
  for (int e = 0; e < kE; ++e) {
    const float* row =
        (e == 0 ? g0 : e == 1 ? g1 : e == 2 ? g2 : g3) + (size_t)b * L;
    float bv = -__builtin_inff();
    int bi = 0x7fffffff;

#if ORACLE_USE_ASYNC
    LDS_AS v4i* ltile = (LDS_AS v4i*)&tile[0][0];
    const float4* grow = (const float4*)row;
    auto issue = [&](int chunk) {
      int idx = chunk * kThreads + tid;
      if (idx < vecRow) {
        // ASYNCcnt-tracked DMA: global -> LDS, 16 B per lane, no VGPR staging
        __builtin_amdgcn_global_load_async_to_lds_b128(
            (GLB_AS v4i*)(uintptr_t)(grow + idx),
            (LDS_AS v4i*)(ltile + (chunk % kDepth) * kThreads + tid),
            /*offset=*/0, /*cpol=*/0);
      }
    };
    for (int p = 0; p < kDepth - 1 && p < nchunks; ++p) issue(p);
    for (int c = 0; c < nchunks; ++c) {
      const int ca = c + kDepth - 1;
      if (ca < nchunks) {
        issue(ca);
        WAIT_ASYNCCNT(kDepth - 1);   // in-order completion => chunk c is in LDS
      } else {
        WAIT_ASYNCCNT(0);            // drain tail (cheap no-op once counter==0)
      }
      asm volatile("" ::: "memory");
      const int idx = c * kThreads + tid;
      if (idx < vecRow) {
        float4 v = tile[c % kDepth][tid];   // ds_load_b128, own slot only
        const int base = idx << 2;
        if (v.x > bv) { bv = v.x; bi = base; }
        if (v.y > bv) { bv = v.y; bi = base + 1; }
        if (v.z > bv) { bv = v.z; bi = base + 2; }
        if (v.w > bv) { bv = v.w; bi = base + 3; }
      }
    }
#else
    const float4* grow = (const float4*)row;
    for (int idx = tid; idx < vecRow; idx += kThreads) {
      float4 v = grow[idx];
      const int base = idx << 2;
      if (v.x > bv) { bv = v.x; bi = base; }
      if (v.y > bv) { bv = v.y; bi = base + 1; }
      if (v.z > bv) { bv = v.z; bi = base + 2; }
      if (v.w > bv) { bv = v.w; bi = base + 3; }
    }
#endif

    // wave32 argmax reduction; ties resolved to the lowest index (jnp.argmax)
#pragma unroll
    for (int off = 16; off > 0; off >>= 1) {
      float ov = __shfl_down(bv, off, 32);
      int   oi = __shfl_down(bi, off, 32);
      if (ov > bv || (ov == bv && oi < bi)) { bv = ov; bi = oi; }
    }
    __syncthreads();                       // red_* reused across experts
    if ((tid & 31) == 0) { red_v[tid >> 5] = bv; red_i[tid >> 5] = bi; }
    __syncthreads();
    if (tid == 0) {
      float fv = red_v[0];
      int fi = red_i[0];
#pragma unroll
      for (int w = 1; w < kWaves; ++w) {
        if (red_v[w] > fv || (red_v[w] == fv && red_i[w] < fi)) {
          fv = red_v[w];
          fi = red_i[w];
        }
      }
      s_pred[e] = fi;
      s_lab[e] = row[label];               // logit at the true label
    }
  }

  __syncthreads();
  if (tid == 0) {
    int best = -1;
#pragma unroll
    for (int e = 0; e < kE; ++e)
      if (best < 0 && s_pred[e] == label) best = e;     // first correct expert
    if (best < 0) {                                     // none correct: conf_best
      float fv = s_lab[0];
      best = 0;
#pragma unroll
      for (int e = 1; e < kE; ++e)
        if (s_lab[e] > fv) { fv = s_lab[e]; best = e; } // strict > => first max
    }
    s_best = best;
  }
  __syncthreads();

  // Emit the chosen expert's row (just streamed by this block => likely L2-warm)
  const int be = s_best;
  const float* brow =
      (be == 0 ? g0 : be == 1 ? g1 : be == 2 ? g2 : g3) + (size_t)b * L;
  const float4* src = (const float4*)brow;
  float4* dst = (float4*)(out + (size_t)b * L);
  for (int idx = tid; idx < vecRow; idx += kThreads) dst[idx] = src[idx];
}

extern "C" void kernel_launch(void* const* d_in, const int* in_sizes, int n_in,
                              void* d_out, int out_size, void* d_ws, size_t ws_size,
                              hipStream_t stream) {
  // setup_inputs() order: labels (int32, B), logits0..3 (float32, B*L each)
  const int*   labels = (const int*)d_in[0];
  const float* g0 = (const float*)d_in[1];
  const float* g1 = (const float*)d_in[2];
  const float* g2 = (const float*)d_in[3];
  const float* g3 = (const float*)d_in[4];
  float* out = (float*)d_out;

  const int B = in_sizes[0];
  const int L = (B > 0) ? in_sizes[1] / B : 0;

  oracle_effective_kernel<<<dim3((unsigned)B), dim3(kThreads), 0, stream>>>(
      g0, g1, g2, g3, labels, out, L);
}